// StyleAttentionExtractor_31078383354206
// MI455X (gfx1250) — compile-verified
//
#include <hip/hip_runtime.h>
#include <hip/hip_bf16.h>

typedef __attribute__((ext_vector_type(2))) float v2f;
typedef __attribute__((ext_vector_type(8))) float v8f;

#define B_     16
#define C_     512
#define HW_    4096     // 64*64
#define S_     19
#define SPAD_  32
#define SEGHW_ 65536    // 256*256

// ---------------------------------------------------------------------------
// Kernel 1: build per-pixel segment bitmask (nearest-resize = sample every 4th)
// bits[b*HW_ + p] bit s  <=>  seg[b, s, 4h, 4w] != 0
// ---------------------------------------------------------------------------
__global__ void k_maskbits(const float* __restrict__ seg, unsigned* __restrict__ bits) {
    int idx = blockIdx.x * blockDim.x + threadIdx.x;  // B_*HW_ threads
    int b = idx >> 12;
    int p = idx & (HW_ - 1);
    int h = p >> 6, w = p & 63;
    const float* sp = seg + (size_t)b * S_ * SEGHW_ + (size_t)(h * 4) * 256 + (w * 4);
    unsigned m = 0;
#pragma unroll
    for (int s = 0; s < S_; ++s) {
        float v = sp[(size_t)s * SEGHW_];
        m |= (v != 0.0f) ? (1u << s) : 0u;
    }
    bits[idx] = m;
}

// ---------------------------------------------------------------------------
// Kernel 2: counts[b,s] = popcount of bit s over pixels (block reduction)
// ---------------------------------------------------------------------------
__global__ void k_counts(const unsigned* __restrict__ bits, float* __restrict__ counts) {
    int b = blockIdx.x;
    int s = blockIdx.y;
    __shared__ int red[256];
    int t = threadIdx.x;
    int acc = 0;
    for (int p = t; p < HW_; p += 256) acc += (int)((bits[b * HW_ + p] >> s) & 1u);
    red[t] = acc;
    __syncthreads();
    for (int off = 128; off > 0; off >>= 1) {
        if (t < off) red[t] += red[t + off];
        __syncthreads();
    }
    if (t == 0) counts[b * SPAD_ + s] = (float)red[0];
}

// ---------------------------------------------------------------------------
// Kernel 3: sums[b, s, c] = sum_p x[b,c,p] * mask[b,s,p]
// 8 waves per block, each covering a disjoint 512-pixel K-chunk with
// v_wmma_f32_16x16x4_f32 accumulators; deterministic LDS combine at the end.
// A (16x4 f32) = x channel tile; B (4x16 f32) = mask bits -> {0,1} floats.
// Two accumulators cover segment columns 0..15 and 16..31 (pads are zero).
// ---------------------------------------------------------------------------
__global__ void k_masked_sums(const float* __restrict__ x, const unsigned* __restrict__ bits,
                              float* __restrict__ sums) {
    int b  = blockIdx.x >> 5;
    int c0 = (blockIdx.x & 31) << 4;
    int wv   = threadIdx.x >> 5;            // wave 0..7 -> K chunk
    int lane = threadIdx.x & 31;
    int n    = lane & 15;                   // A row (channel) and B col (segment)
    int koff = (lane >> 4) << 1;            // 0 or 2 (K phase per half-wave)
    int pbeg = wv * (HW_ / 8);              // 512-pixel chunk

    const float*    xp = x + ((size_t)(b * C_ + c0 + n)) * HW_ + koff;
    const unsigned* bp = bits + (size_t)b * HW_ + koff;

    v8f acc0 = {};
    v8f acc1 = {};
    for (int p0 = pbeg; p0 < pbeg + HW_ / 8; p0 += 4) {
        v2f a;
        a.x = xp[p0];
        a.y = xp[p0 + 1];
        unsigned w0 = bp[p0];
        unsigned w1 = bp[p0 + 1];
        v2f bb0, bb1;
        bb0.x = (float)((w0 >> n) & 1u);
        bb0.y = (float)((w1 >> n) & 1u);
        bb1.x = (float)((w0 >> (n + 16)) & 1u);
        bb1.y = (float)((w1 >> (n + 16)) & 1u);
        acc0 = __builtin_amdgcn_wmma_f32_16x16x4_f32(false, a, false, bb0, (short)0, acc0,
                                                     false, false);
        acc1 = __builtin_amdgcn_wmma_f32_16x16x4_f32(false, a, false, bb1, (short)0, acc1,
                                                     false, false);
    }

    // Deterministic cross-wave combine through LDS.
    __shared__ float lds[8][2][256];
#pragma unroll
    for (int v = 0; v < 8; ++v) {
        lds[wv][0][lane * 8 + v] = acc0[v];
        lds[wv][1][lane * 8 + v] = acc1[v];
    }
    __syncthreads();

    // D layout: VGPR v -> row M = v (+8 for lanes 16..31); lane&15 -> col N
    int t = threadIdx.x;             // 256 positions, one per thread
    int lane_j = t >> 3, v = t & 7;
    int nj = lane_j & 15;
    int ch = c0 + ((lane_j >> 4) << 3) + v;
    float s0 = 0.0f, s1 = 0.0f;
#pragma unroll
    for (int w = 0; w < 8; ++w) {
        s0 += lds[w][0][t];
        s1 += lds[w][1][t];
    }
    sums[((size_t)b * SPAD_ + nj) * C_ + ch]      = s0;
    sums[((size_t)b * SPAD_ + 16 + nj) * C_ + ch] = s1;
}

// ---------------------------------------------------------------------------
// Kernel 4: feat = relu(sums / max(counts,1)) in place
// ---------------------------------------------------------------------------
__global__ void k_feat(float* __restrict__ sums, const float* __restrict__ counts) {
    int idx = blockIdx.x * blockDim.x + threadIdx.x;   // B_*SPAD_*C_
    int bs = idx >> 9;                                 // b*SPAD_ + s
    float cnt = counts[bs];
    float v = sums[idx] / fmaxf(cnt, 1.0f);
    sums[idx] = fmaxf(v, 0.0f);
}

// ---------------------------------------------------------------------------
// Kernel 5: out[b,s,o] = feat[b,s,:] . Ws[s,o,:] + bs[s,o], zero if count==0
// 4 waves per block, each covering a disjoint 128-channel K-chunk:
// M=16 batches, N=16 outputs, K=512; deterministic LDS combine + epilogue.
// ---------------------------------------------------------------------------
__global__ void k_out(const float* __restrict__ feat, const float* __restrict__ Ws,
                      const float* __restrict__ bvec, const float* __restrict__ counts,
                      float* __restrict__ out) {
    int s  = blockIdx.x >> 5;
    int o0 = (blockIdx.x & 31) << 4;
    int wv   = threadIdx.x >> 5;           // wave 0..3 -> K chunk
    int lane = threadIdx.x & 31;
    int n    = lane & 15;                  // A row (batch) / B col (output)
    int koff = (lane >> 4) << 1;
    int kbeg = wv * (C_ / 4);              // 128-channel chunk

    const float* ap = feat + ((size_t)n * SPAD_ + s) * C_ + koff;      // batch = n
    const float* wp = Ws + ((size_t)s * C_ + o0 + n) * C_ + koff;      // output = o0+n

    v8f acc = {};
    for (int k0 = kbeg; k0 < kbeg + C_ / 4; k0 += 4) {
        v2f a, bb;
        a.x  = ap[k0];
        a.y  = ap[k0 + 1];
        bb.x = wp[k0];
        bb.y = wp[k0 + 1];
        acc = __builtin_amdgcn_wmma_f32_16x16x4_f32(false, a, false, bb, (short)0, acc,
                                                    false, false);
    }

    __shared__ float lds[4][256];
#pragma unroll
    for (int v = 0; v < 8; ++v) lds[wv][lane * 8 + v] = acc[v];
    __syncthreads();

    // 128 threads handle 256 D positions (2 each), add bias, mask empty segs.
#pragma unroll
    for (int rep = 0; rep < 2; ++rep) {
        int t2 = threadIdx.x + rep * 128;
        int lane_j = t2 >> 3, v = t2 & 7;
        int batch = ((lane_j >> 4) << 3) + v;
        int o = o0 + (lane_j & 15);
        float val = 0.0f;
#pragma unroll
        for (int w = 0; w < 4; ++w) val += lds[w][t2];
        val += bvec[s * C_ + o];
        float cnt = counts[batch * SPAD_ + s];
        out[((size_t)batch * S_ + s) * C_ + o] = (cnt > 0.0f) ? val : 0.0f;
    }
}

// ---------------------------------------------------------------------------
// Launcher
// ---------------------------------------------------------------------------
extern "C" void kernel_launch(void* const* d_in, const int* in_sizes, int n_in,
                              void* d_out, int out_size, void* d_ws, size_t ws_size,
                              hipStream_t stream) {
    const float* x   = (const float*)d_in[0];  // [16,512,64,64]
    const float* seg = (const float*)d_in[1];  // [16,19,256,256]
    const float* Ws  = (const float*)d_in[2];  // [19,512,512]
    const float* bs  = (const float*)d_in[3];  // [19,512]
    float* out = (float*)d_out;                // [16,19,512]

    // Workspace carve-up (4-byte units):
    //   counts: 512 floats               @ 0
    //   bits:   B_*HW_ = 65536 u32       @ 512
    //   sums:   B_*SPAD_*C_ = 262144 f32 @ 66048
    float*    counts = (float*)d_ws;
    unsigned* bits   = (unsigned*)d_ws + 512;
    float*    sums   = (float*)d_ws + 66048;

    k_maskbits<<<(B_ * HW_) / 256, 256, 0, stream>>>(seg, bits);
    k_counts<<<dim3(B_, S_), 256, 0, stream>>>(bits, counts);
    k_masked_sums<<<B_ * (C_ / 16), 256, 0, stream>>>(x, bits, sums);
    k_feat<<<(B_ * SPAD_ * C_) / 256, 256, 0, stream>>>(sums, counts);
    k_out<<<S_ * (C_ / 16), 128, 0, stream>>>(sums, Ws, bs, counts, out);
}